// MaxPool2dComplex_12317966205093
// MI455X (gfx1250) — compile-verified
//
#include <hip/hip_runtime.h>
#include <stdint.h>

typedef __attribute__((ext_vector_type(4))) float vfloat4;
typedef __attribute__((ext_vector_type(2))) float vfloat2;

namespace {
constexpr int BATCH = 16;
constexpr int CH    = 64;
constexpr int W     = 224;
constexpr int OH    = 112;
constexpr int OW    = 112;
constexpr int PAIRS   = OW / 2;      // 56 float4-strips per output row
constexpr int ROWS_PB = 4;           // output rows per block
constexpr int PLANES  = BATCH * CH;  // 1024 (b,d) planes
constexpr int BLOCK_T = PAIRS * ROWS_PB;           // 224 threads = 7 waves
constexpr size_t PLANE_IN  = (size_t)224 * 224;    // 50176 elems
constexpr size_t PLANE_OUT = (size_t)OH * OW;      // 12544 elems
} // namespace

// No-FMA norm to match the reference's mul,mul,add rounding exactly.
__device__ __forceinline__ float norm2_nc(float r, float i) {
  return __fadd_rn(__fmul_rn(r, r), __fmul_rn(i, i));
}

// GLOBAL_LOAD_ASYNC_TO_LDS_B128, GVS form: per-lane
//   copies 16 bytes from global address sbase + voff to the LDS address in lds_addr, ASYNCcnt-tracked.
// Plane base is wave-uniform -> SGPR pair; per-lane offset is a 32-bit byte
// offset, eliminating 64-bit per-lane address math.
__device__ __forceinline__ void async_g2l_b128(unsigned lds_addr, unsigned voff,
                                               const float* sbase) {
  asm volatile("global_load_async_to_lds_b128 %0, %1, %2"
               :
               : "v"(lds_addr), "v"(voff), "s"(sbase)
               : "memory");
}
__device__ __forceinline__ void wait_async0() {
  asm volatile("s_wait_asynccnt 0" ::: "memory");
}

// argmax over window order (0,0),(0,1),(1,0),(1,1); strict '>' == first-max
// tie-break, matching jnp.argmax.
__device__ __forceinline__ void sel4(float ra, float ia, float rb, float ib,
                                     float rc, float ic2, float rd, float id,
                                     float& orr, float& oi) {
  float best = norm2_nc(ra, ia);
  orr = ra; oi = ia;
  float n = norm2_nc(rb, ib);
  if (n > best) { best = n; orr = rb; oi = ib; }
  n = norm2_nc(rc, ic2);
  if (n > best) { best = n; orr = rc; oi = ic2; }
  n = norm2_nc(rd, id);
  if (n > best) { orr = rd; oi = id; }
}

__global__ __launch_bounds__(BLOCK_T) void
maxpool2d_complex_kernel(const float* __restrict__ x, float* __restrict__ out) {
  const int tx    = (int)threadIdx.x;              // strip id (pw), 0..55
  const int ty    = (int)threadIdx.y;              // local output row, 0..3
  const int plane = (int)blockIdx.y;               // (b,d) plane, 0..1023
  const int oh    = (int)blockIdx.x * ROWS_PB + ty;
  const int b     = plane >> 6;

  // real channel plane = (128b + d) = plane + 64b ; imag = real + 64 planes
  const float* realp = x + (size_t)(plane + (b << 6)) * PLANE_IN;
  const float* imagp = realp + (size_t)CH * PLANE_IN;

  // per-lane byte offsets within the plane: row ih=2*oh, col 4*tx
  const unsigned voff0 = (unsigned)(oh * (2 * W * 4) + tx * 16); // row ih
  const unsigned voff1 = voff0 + (unsigned)(W * 4);              // row ih+1

  __shared__ vfloat4 lbuf[4][ROWS_PB][PAIRS];
  const unsigned l0 = (unsigned)(uintptr_t)&lbuf[0][ty][tx];
  const unsigned l1 = (unsigned)(uintptr_t)&lbuf[1][ty][tx];
  const unsigned l2 = (unsigned)(uintptr_t)&lbuf[2][ty][tx];
  const unsigned l3 = (unsigned)(uintptr_t)&lbuf[3][ty][tx];

  // 4x async copies into this lane's private LDS slots (no barrier needed).
  async_g2l_b128(l0, voff0, realp);
  async_g2l_b128(l1, voff1, realp);
  async_g2l_b128(l2, voff0, imagp);
  async_g2l_b128(l3, voff1, imagp);
  wait_async0();   // hardware + compiler ordering before the ds_loads below

  const vfloat4 r0 = lbuf[0][ty][tx];
  const vfloat4 r1 = lbuf[1][ty][tx];
  const vfloat4 i0 = lbuf[2][ty][tx];
  const vfloat4 i1 = lbuf[3][ty][tx];

  // Window 0: cols {0,1}; Window 1: cols {2,3} of the float4 strip.
  float orA, oiA, orB, oiB;
  sel4(r0.x, i0.x, r0.y, i0.y, r1.x, i1.x, r1.y, i1.y, orA, oiA);
  sel4(r0.z, i0.z, r0.w, i0.w, r1.z, i1.z, r1.w, i1.w, orB, oiB);

  float* orp = out + (size_t)(plane + (b << 6)) * PLANE_OUT
                   + (size_t)oh * OW + tx * 2;
  float* oip = orp + (size_t)CH * PLANE_OUT;

  // Output is written once, never re-read on-chip -> non-temporal stores.
  vfloat2 vr = {orA, orB};
  vfloat2 vi = {oiA, oiB};
  __builtin_nontemporal_store(vr, (vfloat2*)orp);
  __builtin_nontemporal_store(vi, (vfloat2*)oip);
}

extern "C" void kernel_launch(void* const* d_in, const int* in_sizes, int n_in,
                              void* d_out, int out_size, void* d_ws, size_t ws_size,
                              hipStream_t stream) {
  (void)in_sizes; (void)n_in; (void)d_ws; (void)ws_size; (void)out_size;
  const float* x = (const float*)d_in[0];
  float* out = (float*)d_out;
  dim3 grid(OH / ROWS_PB, PLANES);     // (28, 1024), exact cover
  dim3 block(PAIRS, ROWS_PB);          // (56, 4) = 224 threads
  maxpool2d_complex_kernel<<<grid, block, 0, stream>>>(x, out);
}